// ProjChamfer_35115652612619
// MI455X (gfx1250) — compile-verified
//
#include <hip/hip_runtime.h>
#include <hip/hip_bf16.h>

typedef __attribute__((ext_vector_type(2))) float v2f;
typedef __attribute__((ext_vector_type(8))) float v8f;
typedef __attribute__((ext_vector_type(4))) int   v4i;

#define CH_EPS 1e-12f
#define BIGF   3.0e38f
#define INF_BITS 0x7F800000u
#define TILE_ROWS 128
#define MAXM 4096
#define RED_BLOCKS 256

// ---------------------------------------------------------------------------
// Kernel 1: project points to image plane, build K=4 WMMA operand vectors,
// init min buffers.
//   gt  row vector A[m] = [u, v, u^2+v^2, 1]
//   pred col vector B[n] = [-2u, -2v, 1, u^2+v^2]
// so A[m]·B[n] = |a|^2 + |b|^2 - 2 a·b = squared distance.
// ---------------------------------------------------------------------------
__global__ __launch_bounds__(256)
void proj_kernel(const float* __restrict__ pred, const float* __restrict__ gt,
                 const float* __restrict__ P,
                 float4* __restrict__ gtA, float4* __restrict__ predB,
                 float* __restrict__ rowmin, unsigned* __restrict__ colminU,
                 int B, int N, int M) {
    int total = B * (N + M);
    for (int i = blockIdx.x * blockDim.x + threadIdx.x; i < total;
         i += gridDim.x * blockDim.x) {
        bool isGt = (i < B * N);
        int b, idx;
        const float* src;
        if (isGt) {
            b = i / N; idx = i - b * N;
            src = gt + ((size_t)b * N + idx) * 3;
        } else {
            int j = i - B * N;
            b = j / M; idx = j - b * M;
            src = pred + ((size_t)b * M + idx) * 3;
        }
        float x = src[0], y = src[1], z = src[2];
        const float* Pb = P + b * 12;
        float p0 = Pb[0] * x + Pb[1] * y + Pb[2]  * z + Pb[3];
        float p1 = Pb[4] * x + Pb[5] * y + Pb[6]  * z + Pb[7];
        float p2 = Pb[8] * x + Pb[9] * y + Pb[10] * z + Pb[11];
        float u = p0 / p2, v = p1 / p2;
        float nn = u * u + v * v;
        if (isGt) {
            gtA[(size_t)b * N + idx]    = make_float4(u, v, nn, 1.0f);
            rowmin[(size_t)b * N + idx] = BIGF;
        } else {
            predB[(size_t)b * M + idx]   = make_float4(-2.0f * u, -2.0f * v, 1.0f, nn);
            colminU[(size_t)b * M + idx] = INF_BITS;
        }
    }
}

// ---------------------------------------------------------------------------
// Kernel 2: tiled pairwise d^2 via V_WMMA_F32_16X16X4_F32, min-reduced.
// grid = (N/128, B), block = 256 (8 waves). Wave w owns rows
// [blk*128 + w*16, +16) and sweeps all M/16 column tiles.
// predB batch slice (64KB) is staged once per WG into LDS via async
// global->LDS DMA; the inner loop is then 1x ds_load_b64 (pipelined)
// + 1x v_wmma + min-reduce + 1x ds_min_u32.
// ---------------------------------------------------------------------------
__global__ __launch_bounds__(256)
void chamfer_kernel(const float4* __restrict__ gtA, const float4* __restrict__ predB,
                    float* __restrict__ rowmin, unsigned* __restrict__ colminU,
                    int N, int M) {
    __shared__ float4   bstage[MAXM];   // 64 KB: B operand vectors for this batch
    __shared__ unsigned colLds[MAXM];   // 16 KB: per-column running min (bits)

    const int tid = threadIdx.x;
    const int b = blockIdx.y;

    for (int c = tid; c < M; c += 256) colLds[c] = INF_BITS;

    // ---- stage predB[b] into LDS (once per workgroup) ----
    const float4* gsrc = predB + (size_t)b * M;
#if __has_builtin(__builtin_amdgcn_global_load_async_to_lds_b128)
    for (int i = tid; i < M; i += 256) {
        __builtin_amdgcn_global_load_async_to_lds_b128(
            (v4i*)(gsrc + i), (v4i*)(&bstage[i]), 0, 0);
    }
#if __has_builtin(__builtin_amdgcn_s_wait_asynccnt)
    __builtin_amdgcn_s_wait_asynccnt(0);
#else
    asm volatile("s_wait_asynccnt 0x0" ::: "memory");
#endif
#else
    for (int i = tid; i < M; i += 256) bstage[i] = gsrc[i];
#endif
    __syncthreads();

    const int wave = tid >> 5;
    const int lane = tid & 31;
    const int hi   = lane >> 4;        // lane-half: selects K pair, row offset
    const int sub  = lane & 15;        // row (A) / col (B) within tile
    const int kbase = hi * 2;
    const int rowbase = blockIdx.x * TILE_ROWS + wave * 16;

    // A fragment: constant across all column tiles for this wave.
    const float* aptr = (const float*)&gtA[(size_t)b * N + rowbase + sub];
    v2f afrag;
    {
        float2 av = *(const float2*)(aptr + kbase);
        afrag.x = av.x; afrag.y = av.y;
    }

    float rowacc[8];
#pragma unroll
    for (int j = 0; j < 8; ++j) rowacc[j] = BIGF;

    const float* bs = (const float*)&bstage[0];
    const int ntiles = M >> 4;

    // software-pipelined B-fragment fetch from LDS
    v2f bfrag;
    {
        float2 bv = *(const float2*)(bs + (size_t)sub * 4 + kbase);
        bfrag.x = bv.x; bfrag.y = bv.y;
    }

    for (int t = 0; t < ntiles; ++t) {
        const int tn = (t + 1 < ntiles) ? t + 1 : t;
        float2 bnext = *(const float2*)(bs + (size_t)(tn * 16 + sub) * 4 + kbase);

        v8f cz = {};
        // D = A(16x4) x B(4x16): full 16x16 tile of squared distances.
        v8f d = __builtin_amdgcn_wmma_f32_16x16x4_f32(
            false, afrag, false, bfrag, (short)0, cz, false, false);

        float cpart = fminf(d[0], d[1]);
#pragma unroll
        for (int j = 0; j < 8; ++j) rowacc[j] = fminf(rowacc[j], d[j]);
#pragma unroll
        for (int j = 2; j < 8; ++j) cpart = fminf(cpart, d[j]);
        // clamp once (nonneg floats order like uints); both lane-halves
        // atomically combine into the shared per-column min.
        cpart = fmaxf(cpart, 0.0f);
        atomicMin(&colLds[t * 16 + sub], __float_as_uint(cpart)); // ds_min_u32

        bfrag.x = bnext.x; bfrag.y = bnext.y;
    }

    // Row minima: butterfly min across the 16 lanes of each half.
#pragma unroll
    for (int mask = 1; mask <= 8; mask <<= 1) {
#pragma unroll
        for (int j = 0; j < 8; ++j)
            rowacc[j] = fminf(rowacc[j], __shfl_xor(rowacc[j], mask, 32));
    }
    if (sub < 8) {
        float v = rowacc[0];
#pragma unroll
        for (int j = 1; j < 8; ++j)
            if (sub == j) v = rowacc[j];
        // this wave exclusively owns these rows -> plain store
        rowmin[(size_t)b * N + rowbase + 8 * hi + sub] = v;
    }

    __syncthreads();
    for (int c = tid; c < M; c += 256)
        atomicMin(&colminU[(size_t)b * M + c], colLds[c]);  // global_atomic_min_u32
}

// ---------------------------------------------------------------------------
// Kernel 3a/3b: deterministic two-stage reduction of
//   mean_b( mean_row sqrt(max(d2,eps)) + mean_col sqrt(max(d2,eps)) )
// ---------------------------------------------------------------------------
__global__ __launch_bounds__(256)
void reduce_partial(const float* __restrict__ rowmin,
                    const unsigned* __restrict__ colminU,
                    float* __restrict__ partial, int B, int N, int M) {
    __shared__ float sbuf[256];
    const int totalR = B * N, totalC = B * M;
    const int total = totalR + totalC;
    const float wr = 1.0f / ((float)B * (float)N);
    const float wc = 1.0f / ((float)B * (float)M);
    float acc = 0.0f;
    for (int i = blockIdx.x * 256 + threadIdx.x; i < total; i += gridDim.x * 256) {
        float v = (i < totalR) ? rowmin[i] : __uint_as_float(colminU[i - totalR]);
        float w = (i < totalR) ? wr : wc;
        acc += w * sqrtf(fmaxf(v, CH_EPS));
    }
    sbuf[threadIdx.x] = acc;
    __syncthreads();
    for (int s = 128; s > 0; s >>= 1) {
        if (threadIdx.x < s) sbuf[threadIdx.x] += sbuf[threadIdx.x + s];
        __syncthreads();
    }
    if (threadIdx.x == 0) partial[blockIdx.x] = sbuf[0];
}

__global__ __launch_bounds__(256)
void reduce_final(const float* __restrict__ partial, float* __restrict__ out, int n) {
    __shared__ float sbuf[256];
    float acc = 0.0f;
    for (int i = threadIdx.x; i < n; i += 256) acc += partial[i];
    sbuf[threadIdx.x] = acc;
    __syncthreads();
    for (int s = 128; s > 0; s >>= 1) {
        if (threadIdx.x < s) sbuf[threadIdx.x] += sbuf[threadIdx.x + s];
        __syncthreads();
    }
    if (threadIdx.x == 0) out[0] = sbuf[0];
}

// ---------------------------------------------------------------------------
extern "C" void kernel_launch(void* const* d_in, const int* in_sizes, int n_in,
                              void* d_out, int out_size, void* d_ws, size_t ws_size,
                              hipStream_t stream) {
    const float* pred = (const float*)d_in[0];
    const float* gt   = (const float*)d_in[1];
    const float* P    = (const float*)d_in[2];

    const int B = in_sizes[2] / 12;          // 8
    const int N = in_sizes[1] / (3 * B);     // gt points  (rows)   = 4096
    const int M = in_sizes[0] / (3 * B);     // pred points (cols)  = 4096

    // workspace layout (all 16B aligned)
    float*    gtA     = (float*)d_ws;                        // B*N*4 floats
    float*    predB   = gtA   + (size_t)B * N * 4;           // B*M*4 floats
    float*    rowmin  = predB + (size_t)B * M * 4;           // B*N floats
    unsigned* colminU = (unsigned*)(rowmin + (size_t)B * N); // B*M uints
    float*    partial = (float*)(colminU + (size_t)B * M);   // RED_BLOCKS floats

    {
        int total = B * (N + M);
        int blocks = (total + 255) / 256;
        proj_kernel<<<blocks, 256, 0, stream>>>(pred, gt, P,
                                                (float4*)gtA, (float4*)predB,
                                                rowmin, colminU, B, N, M);
    }
    {
        dim3 grid(N / TILE_ROWS, B);
        chamfer_kernel<<<grid, 256, 0, stream>>>((const float4*)gtA,
                                                 (const float4*)predB,
                                                 rowmin, colminU, N, M);
    }
    reduce_partial<<<RED_BLOCKS, 256, 0, stream>>>(rowmin, colminU, partial, B, N, M);
    reduce_final<<<1, 256, 0, stream>>>(partial, (float*)d_out, RED_BLOCKS);
}